// LERS_80719615361507
// MI455X (gfx1250) — compile-verified
//
#include <hip/hip_runtime.h>
#include <math.h>

#define T_  8
#define E_  50000
#define NA_ 4096
#define NL_ 753
#define D_  128
#define H_  8
#define HD_ 16
#define FF_ 2048
#define L_  3

typedef float v2f __attribute__((ext_vector_type(2)));
typedef float v8f __attribute__((ext_vector_type(8)));

// ---------------------------------------------------------------------------
// Generic fp32 WMMA GEMM:  C[b] = act( A[b] (MxK, lda) * W[b]^T (NxK, row-major)
//                                      + bias[b] + Res[b] )
// One wave32 per 32x32 C tile (2x2 register blocking over 16x16 WMMA tiles):
// per K-step of 4 we load 2 A-frags + 2 W-frags (8B each per lane) and issue
// 4 x V_WMMA_F32_16X16X4_F32  -> 8 bytes/lane per wmma instead of 16.
// Frag layouts (ISA 7.12.2):
//   A (16x4):  lane L -> row = L&15, K pair = (L>>4)*2      (2 VGPRs)
//   B (4x16):  lane L -> col = L&15, K pair = (L>>4)*2      (2 VGPRs)
//   C/D:       VGPR r, lane L -> row = r + 8*(L>=16), col = L&15
// ---------------------------------------------------------------------------
__global__ __launch_bounds__(32) void gemm_wmma_f32(
    const float* __restrict__ A, const float* __restrict__ W,
    const float* __restrict__ bias, const float* __restrict__ Res,
    float* __restrict__ C,
    int M, int N, int K, int lda, int ldc, int ldres, int relu,
    long aBS, long wBS, long bBS, long cBS, long rBS)
{
  const int b      = blockIdx.y;
  const int tilesN = (N + 31) >> 5;
  const int tm     = blockIdx.x / tilesN;
  const int tn     = blockIdx.x % tilesN;

  const float* Ab = A + (long)b * aBS;
  const float* Wb = W + (long)b * wBS;
  float*       Cb = C + (long)b * cBS;

  const int lane = threadIdx.x;
  const int half = lane >> 4;     // 0: K 0..1, 1: K 2..3
  const int lidx = lane & 15;
  const int kk   = half * 2;

  // clamped load rows/cols (pad lanes duplicate last row/col; their results
  // land in out-of-range C rows/cols and are never stored)
  int r0 = tm * 32 + lidx;      if (r0 >= M) r0 = M - 1;
  int r1 = tm * 32 + 16 + lidx; if (r1 >= M) r1 = M - 1;
  int c0 = tn * 32 + lidx;      if (c0 >= N) c0 = N - 1;
  int c1 = tn * 32 + 16 + lidx; if (c1 >= N) c1 = N - 1;

  const float* a0p = Ab + (long)r0 * lda + kk;
  const float* a1p = Ab + (long)r1 * lda + kk;
  const float* w0p = Wb + (long)c0 * K   + kk;
  const float* w1p = Wb + (long)c1 * K   + kk;

  v8f acc00 = {}, acc01 = {}, acc10 = {}, acc11 = {};
  for (int k0 = 0; k0 < K; k0 += 4) {
    v2f a0 = *(const v2f*)(a0p + k0);
    v2f a1 = *(const v2f*)(a1p + k0);
    v2f w0 = *(const v2f*)(w0p + k0);
    v2f w1 = *(const v2f*)(w1p + k0);
    acc00 = __builtin_amdgcn_wmma_f32_16x16x4_f32(false, a0, false, w0, (short)0, acc00, false, false);
    acc01 = __builtin_amdgcn_wmma_f32_16x16x4_f32(false, a0, false, w1, (short)0, acc01, false, false);
    acc10 = __builtin_amdgcn_wmma_f32_16x16x4_f32(false, a1, false, w0, (short)0, acc10, false, false);
    acc11 = __builtin_amdgcn_wmma_f32_16x16x4_f32(false, a1, false, w1, (short)0, acc11, false, false);
  }

  const float* Bi = bias ? bias + (long)b * bBS : nullptr;
  const float* Rb = Res  ? Res  + (long)b * rBS : nullptr;
#pragma unroll
  for (int ci = 0; ci < 2; ++ci) {
    const int col = tn * 32 + ci * 16 + lidx;
    if (col >= N) continue;
    const float badd = Bi ? Bi[col] : 0.0f;
#pragma unroll
    for (int ri = 0; ri < 2; ++ri) {
      v8f acc = (ri == 0) ? (ci == 0 ? acc00 : acc01)
                          : (ci == 0 ? acc10 : acc11);
#pragma unroll
      for (int r = 0; r < 8; ++r) {
        const int row = tm * 32 + ri * 16 + half * 8 + r;
        if (row < M) {
          float v = acc[r] + badd;
          if (Rb) v += Rb[(long)row * ldres + col];
          if (relu) v = fmaxf(v, 0.0f);
          Cb[(long)row * ldc + col] = v;
        }
      }
    }
  }
}

// ---------------------------------------------------------------------------
// Input projections
// ---------------------------------------------------------------------------
__global__ __launch_bounds__(256) void proj_adm_k(
    const float* __restrict__ xin, const float* __restrict__ w,
    const float* __restrict__ b, float* __restrict__ out)
{
  int i = blockIdx.x * blockDim.x + threadIdx.x;
  if (i >= NA_ * D_) return;
  int n = i / D_, dd = i % D_;
  float acc = b[dd];
  const float* xr = xin + n * 8;
  const float* wr = w + dd * 8;
#pragma unroll
  for (int k = 0; k < 8; ++k) acc += xr[k] * wr[k];
  out[i] = acc;
}

__global__ __launch_bounds__(256) void proj_lab_k(
    const float* __restrict__ xin, const int* __restrict__ nid,
    const float* __restrict__ emb, const float* __restrict__ w,
    const float* __restrict__ b, float* __restrict__ out)
{
  int i = blockIdx.x * blockDim.x + threadIdx.x;
  if (i >= NL_ * D_) return;
  int n = i / D_, dd = i % D_;
  float acc = b[dd] + xin[n * 2] * w[dd * 2] + xin[n * 2 + 1] * w[dd * 2 + 1];
  acc += emb[(long)nid[n] * D_ + dd];
  out[i] = acc;
}

__global__ __launch_bounds__(256) void bcast_k(
    const float* __restrict__ src, float* __restrict__ dst, long per, int reps)
{
  long i = (long)blockIdx.x * blockDim.x + threadIdx.x;
  if (i >= per * (long)reps) return;
  dst[i] = src[i % per];
}

// ---------------------------------------------------------------------------
// GNN: edge messages + segment-sum (both directions fused), one wave per
// (t, edge), lanes parallel over feature dim.  relu(msg)==0 skips the atomic.
// ---------------------------------------------------------------------------
__global__ __launch_bounds__(256) void gnn_edge_k(
    const int* __restrict__ eidx, const float* __restrict__ eattr,
    const float* __restrict__ ew, const float* __restrict__ eb,
    const float* __restrict__ adm, const float* __restrict__ lab,
    float* __restrict__ aggA, float* __restrict__ aggL, int l)
{
  long gid  = (long)blockIdx.x * blockDim.x + threadIdx.x;
  long widx = gid >> 5;
  int  lane = (int)(gid & 31);
  if (widx >= (long)T_ * E_) return;
  int t = (int)(widx / E_);
  int e = (int)(widx % E_);

  int src = eidx[((long)t * 2 + 0) * E_ + e];
  int dst = eidx[((long)t * 2 + 1) * E_ + e];
  float ea0 = eattr[((long)t * E_ + e) * 2 + 0];
  float ea1 = eattr[((long)t * E_ + e) * 2 + 1];

  const float* wl = ew + (((long)t * 2 + l) * 2 + 0) * D_ * 2;
  const float* wa = ew + (((long)t * 2 + l) * 2 + 1) * D_ * 2;
  const float* bl = eb + (((long)t * 2 + l) * 2 + 0) * D_;
  const float* ba = eb + (((long)t * 2 + l) * 2 + 1) * D_;

  const float* asrc = adm + ((long)t * NA_ + src) * D_;
  const float* ldst = lab + ((long)t * NL_ + dst) * D_;
  float* gl = aggL + ((long)t * NL_ + dst) * D_;
  float* ga = aggA + ((long)t * NA_ + src) * D_;

#pragma unroll
  for (int j = 0; j < D_ / 32; ++j) {
    int dd = lane + 32 * j;
    float m1 = asrc[dd] + ea0 * wl[dd * 2] + ea1 * wl[dd * 2 + 1] + bl[dd];
    if (m1 > 0.0f) atomicAdd(&gl[dd], m1);
    float m2 = ldst[dd] + ea0 * wa[dd * 2] + ea1 * wa[dd * 2 + 1] + ba[dd];
    if (m2 > 0.0f) atomicAdd(&ga[dd], m2);
  }
}

// pre = (1+eps[t]) * state + agg   (GIN aggregation input to node MLP)
__global__ __launch_bounds__(256) void gin_pre_k(
    const float* __restrict__ st, const float* __restrict__ agg,
    const float* __restrict__ eps, float* __restrict__ pre,
    int Nn, int l, int dir)
{
  long i = (long)blockIdx.x * blockDim.x + threadIdx.x;
  long total = (long)T_ * Nn * D_;
  if (i >= total) return;
  int t = (int)(i / ((long)Nn * D_));
  float ep = eps[((long)t * 2 + l) * 2 + dir];
  pre[i] = (1.0f + ep) * st[i] + agg[i];
}

// ---------------------------------------------------------------------------
// Causal attention over T=8, head dim 16: one thread per (tq, n, h).
// qkv rows: (t*N + n) * 384, segments [q|k|v] of 128 each.
// ---------------------------------------------------------------------------
__global__ __launch_bounds__(256) void attn_k(
    const float* __restrict__ qkv, float* __restrict__ outb, int N)
{
  long i = (long)blockIdx.x * blockDim.x + threadIdx.x;
  long total = (long)T_ * N * H_;
  if (i >= total) return;
  int h = (int)(i % H_);
  long tn = i / H_;
  int n  = (int)(tn % N);
  int tq = (int)(tn / N);

  float q[HD_];
  const float* qp = qkv + ((long)tq * N + n) * (3 * D_) + h * HD_;
#pragma unroll
  for (int j = 0; j < HD_; ++j) q[j] = qp[j];

  float s[T_];
  float mx = -3.0e38f;
#pragma unroll
  for (int tk = 0; tk < T_; ++tk) {
    const float* kp = qkv + ((long)tk * N + n) * (3 * D_) + D_ + h * HD_;
    float acc = 0.0f;
#pragma unroll
    for (int j = 0; j < HD_; ++j) acc += q[j] * kp[j];
    float sc = acc * 0.25f;              // 1/sqrt(16)
    if (tk > tq) sc = -1.0e9f;           // causal mask
    s[tk] = sc;
    mx = fmaxf(mx, sc);
  }
  float den = 0.0f;
#pragma unroll
  for (int tk = 0; tk < T_; ++tk) { s[tk] = expf(s[tk] - mx); den += s[tk]; }
  float inv = 1.0f / den;

  float o[HD_];
#pragma unroll
  for (int j = 0; j < HD_; ++j) o[j] = 0.0f;
#pragma unroll
  for (int tk = 0; tk < T_; ++tk) {
    const float* vp = qkv + ((long)tk * N + n) * (3 * D_) + 2 * D_ + h * HD_;
    float a = s[tk] * inv;
#pragma unroll
    for (int j = 0; j < HD_; ++j) o[j] += a * vp[j];
  }
  float* op = outb + ((long)tq * N + n) * D_ + h * HD_;
#pragma unroll
  for (int j = 0; j < HD_; ++j) op[j] = o[j];
}

// LayerNorm over D=128: one wave32 per row, shuffle reduction.
__global__ __launch_bounds__(256) void ln_k(
    const float* __restrict__ y, float* __restrict__ x,
    const float* __restrict__ g, const float* __restrict__ bt, long rows)
{
  long gid = (long)blockIdx.x * blockDim.x + threadIdx.x;
  long w   = gid >> 5;
  int lane = (int)(gid & 31);
  if (w >= rows) return;
  const float* yr = y + w * D_;
  float v[4];
  float s = 0.0f, s2 = 0.0f;
#pragma unroll
  for (int j = 0; j < 4; ++j) { v[j] = yr[lane + 32 * j]; s += v[j]; s2 += v[j] * v[j]; }
#pragma unroll
  for (int off = 16; off > 0; off >>= 1) {
    s  += __shfl_xor(s,  off, 32);
    s2 += __shfl_xor(s2, off, 32);
  }
  float m   = s  * (1.0f / D_);
  float var = s2 * (1.0f / D_) - m * m;
  float inv = 1.0f / sqrtf(var + 1e-5f);
  float* xr = x + w * D_;
#pragma unroll
  for (int j = 0; j < 4; ++j) {
    int c = lane + 32 * j;
    xr[c] = (v[j] - m) * inv * g[c] + bt[c];
  }
}

// labels[t][src][dst] = 1  from edge_index[t+1]
__global__ __launch_bounds__(256) void labels_k(
    const int* __restrict__ eidx, float* __restrict__ lbl)
{
  long i = (long)blockIdx.x * blockDim.x + threadIdx.x;
  if (i >= (long)T_ * E_) return;
  int t = (int)(i / E_);
  int e = (int)(i % E_);
  int a = eidx[((long)(t + 1) * 2 + 0) * E_ + e];
  int b = eidx[((long)(t + 1) * 2 + 1) * E_ + e];
  lbl[((long)t * NA_ + a) * NL_ + b] = 1.0f;
}

// ---------------------------------------------------------------------------
// Host side
// ---------------------------------------------------------------------------
static inline int cdivl(long a, long b) { return (int)((a + b - 1) / b); }

static inline void launch_gemm(hipStream_t st, const float* A, const float* W,
    const float* bias, const float* Res, float* C, int M, int N, int K,
    int lda, int ldc, int ldres, int relu, int batch,
    long aBS, long wBS, long bBS, long cBS, long rBS)
{
  int tilesM = (M + 31) / 32, tilesN = (N + 31) / 32;
  dim3 grid((unsigned)(tilesM * tilesN), (unsigned)batch, 1);
  gemm_wmma_f32<<<grid, dim3(32, 1, 1), 0, st>>>(
      A, W, bias, Res, C, M, N, K, lda, ldc, ldres, relu,
      aBS, wBS, bBS, cBS, rBS);
}

extern "C" void kernel_launch(void* const* d_in, const int* in_sizes, int n_in,
                              void* d_out, int out_size, void* d_ws, size_t ws_size,
                              hipStream_t stream)
{
  (void)in_sizes; (void)n_in; (void)out_size; (void)ws_size;

  const float* adm_x       = (const float*)d_in[0];
  const float* lab_x       = (const float*)d_in[1];
  const int*   lab_node_id = (const int*)  d_in[2];
  const int*   edge_index  = (const int*)  d_in[3];
  const float* edge_attr   = (const float*)d_in[4];
  const float* lab_emb     = (const float*)d_in[5];
  const float* admproj_w   = (const float*)d_in[6];
  const float* admproj_b   = (const float*)d_in[7];
  const float* labproj_w   = (const float*)d_in[8];
  const float* labproj_b   = (const float*)d_in[9];
  const float* gnn_nn_w    = (const float*)d_in[10];
  const float* gnn_nn_b    = (const float*)d_in[11];
  const float* gnn_edge_w  = (const float*)d_in[12];
  const float* gnn_edge_b  = (const float*)d_in[13];
  const float* gnn_eps     = (const float*)d_in[14];
  const float* dec_sa_in_w = (const float*)d_in[15];
  const float* dec_sa_in_b = (const float*)d_in[16];
  const float* dec_sa_out_w= (const float*)d_in[17];
  const float* dec_sa_out_b= (const float*)d_in[18];
  const float* dec_ca_in_w = (const float*)d_in[19];
  const float* dec_ca_in_b = (const float*)d_in[20];
  const float* dec_ca_out_w= (const float*)d_in[21];
  const float* dec_ca_out_b= (const float*)d_in[22];
  const float* dec_ff1_w   = (const float*)d_in[23];
  const float* dec_ff1_b   = (const float*)d_in[24];
  const float* dec_ff2_w   = (const float*)d_in[25];
  const float* dec_ff2_b   = (const float*)d_in[26];
  const float* dec_ln_g    = (const float*)d_in[27];
  const float* dec_ln_b    = (const float*)d_in[28];

  // Workspace carve-up (~253 MB of fp32) -----------------------------------
  float* ws = (float*)d_ws;
  size_t off = 0;
  auto take = [&](size_t n) { float* p = ws + off; off += n; return p; };
  float* adm0    = take((size_t)NA_ * D_);
  float* lab0    = take((size_t)NL_ * D_);
  float* stA_adm = take((size_t)T_ * NA_ * D_);
  float* stB_adm = take((size_t)T_ * NA_ * D_);
  float* stA_lab = take((size_t)T_ * NL_ * D_);
  float* stB_lab = take((size_t)T_ * NL_ * D_);
  float* agg_adm = take((size_t)T_ * NA_ * D_);
  float* agg_lab = take((size_t)T_ * NL_ * D_);
  float* pre_adm = take((size_t)T_ * NA_ * D_);
  float* pre_lab = take((size_t)T_ * NL_ * D_);
  float* mem_adm = take((size_t)T_ * NA_ * D_);
  float* mem_lab = take((size_t)T_ * NL_ * D_);
  float* qkv     = take((size_t)T_ * NA_ * 3 * D_);
  float* attno   = take((size_t)T_ * NA_ * D_);
  float* yb      = take((size_t)T_ * NA_ * D_);
  const long FFCH = 8192;                  // FF row chunk
  float* h1      = take((size_t)FFCH * FF_);

  const int thr = 256;

  // 1. Input projections ----------------------------------------------------
  proj_adm_k<<<cdivl((long)NA_ * D_, thr), thr, 0, stream>>>(adm_x, admproj_w, admproj_b, adm0);
  proj_lab_k<<<cdivl((long)NL_ * D_, thr), thr, 0, stream>>>(lab_x, lab_node_id, lab_emb,
                                                             labproj_w, labproj_b, lab0);
  bcast_k<<<cdivl((long)T_ * NA_ * D_, thr), thr, 0, stream>>>(adm0, stA_adm, (long)NA_ * D_, T_);
  bcast_k<<<cdivl((long)T_ * NL_ * D_, thr), thr, 0, stream>>>(lab0, stA_lab, (long)NL_ * D_, T_);

  // 2. GNN (2 GIN layers, both bipartite directions, batched over T) --------
  float* curA = stA_adm; float* nxtA = stB_adm;
  float* curL = stA_lab; float* nxtL = stB_lab;
  for (int l = 0; l < 2; ++l) {
    hipMemsetAsync(agg_adm, 0, sizeof(float) * (size_t)T_ * NA_ * D_, stream);
    hipMemsetAsync(agg_lab, 0, sizeof(float) * (size_t)T_ * NL_ * D_, stream);
    gnn_edge_k<<<cdivl((long)T_ * E_ * 32, thr), thr, 0, stream>>>(
        edge_index, edge_attr, gnn_edge_w, gnn_edge_b, curA, curL, agg_adm, agg_lab, l);
    gin_pre_k<<<cdivl((long)T_ * NL_ * D_, thr), thr, 0, stream>>>(curL, agg_lab, gnn_eps, pre_lab, NL_, l, 0);
    gin_pre_k<<<cdivl((long)T_ * NA_ * D_, thr), thr, 0, stream>>>(curA, agg_adm, gnn_eps, pre_adm, NA_, l, 1);
    // node MLPs with per-t weights: gnn_nn_w layout (T,2,2,D,D)
    launch_gemm(stream, pre_lab, gnn_nn_w + ((long)l * 2 + 0) * D_ * D_,
                gnn_nn_b + ((long)l * 2 + 0) * D_, nullptr, nxtL,
                NL_, D_, D_, D_, D_, 0, /*relu=*/1, T_,
                (long)NL_ * D_, (long)4 * D_ * D_, (long)4 * D_, (long)NL_ * D_, 0);
    launch_gemm(stream, pre_adm, gnn_nn_w + ((long)l * 2 + 1) * D_ * D_,
                gnn_nn_b + ((long)l * 2 + 1) * D_, nullptr, nxtA,
                NA_, D_, D_, D_, D_, 0, /*relu=*/1, T_,
                (long)NA_ * D_, (long)4 * D_ * D_, (long)4 * D_, (long)NA_ * D_, 0);
    float* t0 = curA; curA = nxtA; nxtA = t0;
    float* t1 = curL; curL = nxtL; nxtL = t1;
  }

  // 3. Transformer decoders (d=0: adm, d=1: lab) ----------------------------
  for (int d = 0; d < 2; ++d) {
    const int N   = d ? NL_ : NA_;
    const long rows = (long)T_ * N;
    float* x   = d ? curL : curA;
    float* mem = d ? mem_lab : mem_adm;
    hipMemcpyAsync(mem, x, sizeof(float) * rows * D_, hipMemcpyDeviceToDevice, stream);

    for (int l = 0; l < L_; ++l) {
      const long wo = (long)d * L_ + l;
      const float* saw = dec_sa_in_w  + wo * 3 * D_ * D_;
      const float* sab = dec_sa_in_b  + wo * 3 * D_;
      const float* sow = dec_sa_out_w + wo * D_ * D_;
      const float* sob = dec_sa_out_b + wo * D_;
      const float* caw = dec_ca_in_w  + wo * 3 * D_ * D_;
      const float* cab = dec_ca_in_b  + wo * 3 * D_;
      const float* cow = dec_ca_out_w + wo * D_ * D_;
      const float* cob = dec_ca_out_b + wo * D_;
      const float* f1w = dec_ff1_w + wo * FF_ * D_;
      const float* f1b = dec_ff1_b + wo * FF_;
      const float* f2w = dec_ff2_w + wo * D_ * FF_;
      const float* f2b = dec_ff2_b + wo * D_;
      const float* lg0 = dec_ln_g + (wo * 3 + 0) * D_; const float* lb0 = dec_ln_b + (wo * 3 + 0) * D_;
      const float* lg1 = dec_ln_g + (wo * 3 + 1) * D_; const float* lb1 = dec_ln_b + (wo * 3 + 1) * D_;
      const float* lg2 = dec_ln_g + (wo * 3 + 2) * D_; const float* lb2 = dec_ln_b + (wo * 3 + 2) * D_;

      // --- self-attention: qkv proj -> attn -> out proj (+res) -> LN
      launch_gemm(stream, x, saw, sab, nullptr, qkv,
                  (int)rows, 3 * D_, D_, D_, 3 * D_, 0, 0, 1, 0, 0, 0, 0, 0);
      attn_k<<<cdivl(rows * H_, thr), thr, 0, stream>>>(qkv, attno, N);
      launch_gemm(stream, attno, sow, sob, x, yb,
                  (int)rows, D_, D_, D_, D_, D_, 0, 1, 0, 0, 0, 0, 0);
      ln_k<<<cdivl(rows * 32, thr), thr, 0, stream>>>(yb, x, lg0, lb0, rows);

      // --- cross-attention: q from x, k/v from mem
      launch_gemm(stream, x, caw, cab, nullptr, qkv,
                  (int)rows, D_, D_, D_, 3 * D_, 0, 0, 1, 0, 0, 0, 0, 0);
      launch_gemm(stream, mem, caw + (long)D_ * D_, cab + D_, nullptr, qkv + D_,
                  (int)rows, 2 * D_, D_, D_, 3 * D_, 0, 0, 1, 0, 0, 0, 0, 0);
      attn_k<<<cdivl(rows * H_, thr), thr, 0, stream>>>(qkv, attno, N);
      launch_gemm(stream, attno, cow, cob, x, yb,
                  (int)rows, D_, D_, D_, D_, D_, 0, 1, 0, 0, 0, 0, 0);
      ln_k<<<cdivl(rows * 32, thr), thr, 0, stream>>>(yb, x, lg1, lb1, rows);

      // --- feed-forward, chunked over rows to bound the FF_ hidden buffer
      for (long r0 = 0; r0 < rows; r0 += FFCH) {
        int mr = (int)((rows - r0) < FFCH ? (rows - r0) : FFCH);
        launch_gemm(stream, x + r0 * D_, f1w, f1b, nullptr, h1,
                    mr, FF_, D_, D_, FF_, 0, /*relu=*/1, 1, 0, 0, 0, 0, 0);
        launch_gemm(stream, h1, f2w, f2b, x + r0 * D_, yb + r0 * D_,
                    mr, D_, FF_, FF_, D_, D_, 0, 1, 0, 0, 0, 0, 0);
      }
      ln_k<<<cdivl(rows * 32, thr), thr, 0, stream>>>(yb, x, lg2, lb2, rows);
    }
  }

  // 4. scores[t] = adm_dec[t] @ lab_dec[t]^T   (batched over T)
  float* scores = (float*)d_out;
  float* lbl    = scores + (size_t)T_ * NA_ * NL_;
  launch_gemm(stream, curA, curL, nullptr, nullptr, scores,
              NA_, NL_, D_, D_, NL_, 0, 0, T_,
              (long)NA_ * D_, (long)NL_ * D_, 0, (long)NA_ * NL_, 0);

  // 5. labels: zero then scatter ones
  hipMemsetAsync(lbl, 0, sizeof(float) * (size_t)T_ * NA_ * NL_, stream);
  labels_k<<<cdivl((long)T_ * E_, thr), thr, 0, stream>>>(edge_index, lbl);
}